// MeanLayer_19095424598782
// MI455X (gfx1250) — compile-verified
//
#include <hip/hip_runtime.h>
#include <cstdint>
#include <cstddef>

// Problem constants (match reference)
#define N_NODES 100000
#define N_EDGES 1600000
#define D_FEAT  32

#define EPB     32    // edges per block
#define THREADS 256   // 8 threads per edge (each thread: one float4 quad)

typedef __attribute__((address_space(3))) int lds_int_t;

// ---------------------------------------------------------------------------
// Kernel 1: zero the accumulators (harness poisons d_out / d_ws with 0xAA).
// ---------------------------------------------------------------------------
__global__ void graphmean_zero_kernel(float* __restrict__ out,
                                      float* __restrict__ deg) {
    const int tid = blockIdx.x * blockDim.x + threadIdx.x;
    const int total4 = (N_NODES * D_FEAT) / 4;   // 800000 float4 stores
    if (tid < total4) {
        ((float4*)out)[tid] = make_float4(0.f, 0.f, 0.f, 0.f);
    }
    if (tid < N_NODES) {
        deg[tid] = 0.f;
    }
}

// ---------------------------------------------------------------------------
// Kernel 2: edge scatter. 32 edges per 256-thread block.
//   - Wave 0 stages the 32 src + 32 dst indices into LDS with the CDNA5
//     async global->LDS path (ASYNCcnt), then the block barriers.
//   - Each group of 8 lanes handles one edge: coalesced 128B row gather
//     (global_load_b128 per lane) + 4x global_atomic_add_f32 scatter.
// ---------------------------------------------------------------------------
__global__ __launch_bounds__(THREADS)
void graphmean_scatter_kernel(const float* __restrict__ h,
                              const int*   __restrict__ src,
                              const int*   __restrict__ dst,
                              float*       __restrict__ out,
                              float*       __restrict__ deg) {
    __shared__ int s_idx[2 * EPB];   // [0..31] = src, [32..63] = dst

    const int base = blockIdx.x * EPB;

    if (threadIdx.x < 32) {
        // Raw LDS byte offsets of the staging buffer (generic -> AS3 cast
        // recovers the workgroup-relative LDS offset).
        const unsigned lds0 = (unsigned)(uintptr_t)(lds_int_t*)s_idx;
        const unsigned lsrc = lds0 + threadIdx.x * 4u;
        const unsigned ldst = lds0 + (EPB + threadIdx.x) * 4u;
        const int* gsrc = src + base + threadIdx.x;
        const int* gdst = dst + base + threadIdx.x;
        // CDNA5 async copy: per-lane global address -> per-lane LDS address.
        asm volatile("global_load_async_to_lds_b32 %0, %1, off"
                     :: "v"(lsrc), "v"(gsrc) : "memory");
        asm volatile("global_load_async_to_lds_b32 %0, %1, off"
                     :: "v"(ldst), "v"(gdst) : "memory");
        asm volatile("s_wait_asynccnt 0" ::: "memory");
    }
    __syncthreads();

    const int e = threadIdx.x >> 3;   // edge within block: 0..31
    const int q = threadIdx.x & 7;    // feature quad:      0..7

    const int s = s_idx[e];
    const int d = s_idx[EPB + e];

    // 128B per edge, 16B per lane, contiguous across the 8 lanes of an edge.
    const float4 v = *(const float4*)(h + (size_t)s * D_FEAT + q * 4);

    float* o = out + (size_t)d * D_FEAT + q * 4;
    atomicAdd(o + 0, v.x);
    atomicAdd(o + 1, v.y);
    atomicAdd(o + 2, v.z);
    atomicAdd(o + 3, v.w);

    if (q == 0) {
        atomicAdd(deg + d, 1.0f);
    }
}

// ---------------------------------------------------------------------------
// Kernel 3: normalize in place: out[n,:] *= 1 / max(deg[n], 1).
// ---------------------------------------------------------------------------
__global__ void graphmean_norm_kernel(float* __restrict__ out,
                                      const float* __restrict__ deg) {
    const int tid = blockIdx.x * blockDim.x + threadIdx.x;
    if (tid >= N_NODES * 8) return;
    const int node = tid >> 3;
    const int q    = tid & 7;
    const float scale = 1.0f / fmaxf(deg[node], 1.0f);
    float4* p = (float4*)(out + (size_t)node * D_FEAT + q * 4);
    float4 v = *p;
    v.x *= scale; v.y *= scale; v.z *= scale; v.w *= scale;
    *p = v;
}

// ---------------------------------------------------------------------------
// Launch
// ---------------------------------------------------------------------------
extern "C" void kernel_launch(void* const* d_in, const int* in_sizes, int n_in,
                              void* d_out, int out_size, void* d_ws, size_t ws_size,
                              hipStream_t stream) {
    (void)in_sizes; (void)n_in; (void)out_size; (void)ws_size;

    const float* h   = (const float*)d_in[0];
    const int*   src = (const int*)  d_in[1];
    const int*   dst = (const int*)  d_in[2];
    float* out = (float*)d_out;
    float* deg = (float*)d_ws;   // N_NODES floats of scratch

    const int zthreads = N_NODES * D_FEAT / 4;                 // 800000
    graphmean_zero_kernel<<<(zthreads + 255) / 256, 256, 0, stream>>>(out, deg);

    graphmean_scatter_kernel<<<N_EDGES / EPB, THREADS, 0, stream>>>(
        h, src, dst, out, deg);

    graphmean_norm_kernel<<<(N_NODES * 8 + 255) / 256, 256, 0, stream>>>(out, deg);
}